// EuclideanCodebook_63763084476532
// MI455X (gfx1250) — compile-verified
//
#include <hip/hip_runtime.h>
#include <hip/hip_bf16.h>

typedef __attribute__((ext_vector_type(2))) float v2f;
typedef __attribute__((ext_vector_type(8))) float v8f;

#define N_ROWS 16384
#define DIM 256
#define CODES 8192
#define DECAYC 0.8f
#define OMD 0.2f
#define EPSC 1e-5f
#define MT 128           // rows per workgroup in argmax kernel
#define CT 32            // codes per LDS tile
#define LDS_STRIDE 260   // padded row stride (floats), multiple of 4 -> 16B-aligned b128, bank-conflict-free
#define NTILES (CODES / CT)

// ---------------------------------------------------------------------------
// ||e||^2 per code: one wave per code row
// ---------------------------------------------------------------------------
__global__ void k_enorm(const float* __restrict__ embed, float* __restrict__ enorm) {
    const int wave = (blockIdx.x * blockDim.x + threadIdx.x) >> 5;
    const int lane = threadIdx.x & 31;
    const float* row = embed + (size_t)wave * DIM;
    float s = 0.0f;
    #pragma unroll
    for (int c = 0; c < DIM / 32; ++c) {
        const float v = row[lane + c * 32];
        s += v * v;
    }
    #pragma unroll
    for (int m = 16; m >= 1; m >>= 1) s += __shfl_xor(s, m, 32);
    if (lane == 0) enorm[wave] = s;
}

// ---------------------------------------------------------------------------
// Initialize EMA output regions: cs_new = 0.8*cs ; avg_new = 0.8*avg
// ---------------------------------------------------------------------------
__global__ void k_init(const float* __restrict__ cs, const float* __restrict__ avg,
                       float* __restrict__ cs_new, float* __restrict__ avg_new) {
    const size_t i = (size_t)blockIdx.x * 256 + threadIdx.x;
    avg_new[i] = DECAYC * avg[i];
    if (i < CODES) cs_new[i] = DECAYC * cs[i];
}

// ---------------------------------------------------------------------------
// Async global -> LDS copy of one E tile (CT x DIM floats) via the gfx1250
// ASYNCcnt path: no VGPR staging, tracked separately from LOADcnt.
// Each of 256 threads issues (CT*DIM)/(256*4) = 8 b128 transfers.
// ---------------------------------------------------------------------------
__device__ __forceinline__ void async_copy_tile(const float* __restrict__ gsrc,
                                                const float* lbase, int tid) {
    #pragma unroll
    for (int i = 0; i < (CT * DIM) / (256 * 4); ++i) {
        const int slot = tid + i * 256;       // float4 slot id
        const int r    = slot >> 6;           // 64 float4 per row
        const int c4   = slot & 63;
        const float*   src = gsrc + (size_t)r * DIM + (c4 << 2);
        // generic pointer to __shared__: low 32 bits == LDS byte offset
        const unsigned dst = (unsigned)(uintptr_t)(lbase + r * LDS_STRIDE + (c4 << 2));
        asm volatile("global_load_async_to_lds_b128 %0, %1, off"
                     :: "v"(dst), "v"(src) : "memory");
    }
}

__device__ __forceinline__ void wait_async0() {
    asm volatile("s_wait_asynccnt 0x0" ::: "memory");
}

// ---------------------------------------------------------------------------
// Fused fp32-WMMA score GEMM + running argmax, double-buffered async E tiles.
// Each wave owns 16 rows; A fragments live in registers for all of K=256.
// score = 2*x.e - ||e||^2 (row-constant ||x||^2 dropped).
// ---------------------------------------------------------------------------
__global__ __launch_bounds__(256)
void k_argmax(const float* __restrict__ x, const float* __restrict__ embed,
              const float* __restrict__ enorm, int* __restrict__ idx_out,
              float* __restrict__ ind_out) {
    __shared__ float Es[2 * CT * LDS_STRIDE];   // 66.5 KB, double buffer

    const int tid  = threadIdx.x;
    const int lane = tid & 31;
    const int wave = tid >> 5;
    const int l15  = lane & 15;
    const int h    = lane >> 4;
    const int rowBase = blockIdx.x * MT + wave * 16;

    // A fragment registers: areg[2*kk+j] = X[rowBase+l15][4*kk + 2*h + j]
    float areg[128];
    {
        const float* xrow = x + (size_t)(rowBase + l15) * DIM + 2 * h;
        #pragma unroll
        for (int kk = 0; kk < 64; ++kk) {
            const float2 v = *(const float2*)(xrow + 4 * kk);
            areg[2 * kk]     = v.x;
            areg[2 * kk + 1] = v.y;
        }
    }

    float best[8];
    int   bidx[8];
    #pragma unroll
    for (int v = 0; v < 8; ++v) { best[v] = -3.4e38f; bidx[v] = 0; }

    // prologue: fetch tile 0 into buffer 0
    async_copy_tile(embed, Es, tid);
    wait_async0();
    __syncthreads();

    for (int t = 0; t < NTILES; ++t) {
        const int   cur   = t & 1;
        const bool  more  = (t + 1) < NTILES;
        const float* curb = Es + cur * (CT * LDS_STRIDE);

        // kick off next tile into the other buffer; overlaps with the WMMAs below
        if (more)
            async_copy_tile(embed + (size_t)(t + 1) * CT * DIM,
                            Es + (cur ^ 1) * (CT * LDS_STRIDE), tid);

        const int tileBase = t * CT;
        #pragma unroll
        for (int cb = 0; cb < CT / 16; ++cb) {
            v8f acc = {};
            const float* brow = curb + (cb * 16 + l15) * LDS_STRIDE + 2 * h;
            #pragma unroll
            for (int kk = 0; kk < 64; ++kk) {
                v2f a; a.x = areg[2 * kk]; a.y = areg[2 * kk + 1];
                const float2 bv = *(const float2*)(brow + 4 * kk);
                v2f b; b.x = bv.x; b.y = bv.y;
                acc = __builtin_amdgcn_wmma_f32_16x16x4_f32(
                    /*neg_a=*/false, a, /*neg_b=*/false, b,
                    /*c_mod=*/(short)0, acc, /*reuse_a=*/false, /*reuse_b=*/false);
            }
            const int   code = tileBase + cb * 16 + l15;
            const float en   = enorm[code];
            #pragma unroll
            for (int v = 0; v < 8; ++v) {
                const float s = 2.0f * acc[v] - en;
                const bool  u = s > best[v];   // strict > keeps first (lowest) index
                best[v] = u ? s : best[v];
                bidx[v] = u ? code : bidx[v];
            }
        }

        if (more) wait_async0();   // next tile landed in LDS
        __syncthreads();           // everyone done with cur, next tile visible to all
    }

    // reduce across the 16 lanes that share each row (wave32 halves)
    #pragma unroll
    for (int v = 0; v < 8; ++v) {
        float bv = best[v];
        int   bi = bidx[v];
        #pragma unroll
        for (int m = 1; m < 16; m <<= 1) {
            const float ov = __shfl_xor(bv, m, 32);
            const int   oi = __shfl_xor(bi, m, 32);
            if (ov > bv || (ov == bv && oi < bi)) { bv = ov; bi = oi; }
        }
        if (l15 == 0) {
            const int row = rowBase + v + 8 * h;
            idx_out[row] = bi;
            ind_out[row] = (float)bi;
        }
    }
}

// ---------------------------------------------------------------------------
// Per-row gather (quantize) + scatter of 0.2*x into avg_new and 0.2 into cs_new
// ---------------------------------------------------------------------------
__global__ void k_scatter(const float* __restrict__ x, const float* __restrict__ embed,
                          const int* __restrict__ idx, float* __restrict__ quant,
                          float* __restrict__ cs_new, float* __restrict__ avg_new) {
    const int row  = blockIdx.x;
    const int d    = threadIdx.x;
    const int code = idx[row];
    quant[(size_t)row * DIM + d] = embed[(size_t)code * DIM + d];
    atomicAdd(&avg_new[(size_t)code * DIM + d], OMD * x[(size_t)row * DIM + d]);
    if (d == 0) atomicAdd(&cs_new[code], OMD);
}

// ---------------------------------------------------------------------------
// Single-block sum of cluster_size_new -> S (deterministic)
// ---------------------------------------------------------------------------
__global__ void k_sum(const float* __restrict__ cs_new, float* __restrict__ S) {
    float s = 0.0f;
    for (int i = threadIdx.x; i < CODES; i += 256) s += cs_new[i];
    #pragma unroll
    for (int m = 16; m >= 1; m >>= 1) s += __shfl_xor(s, m, 32);
    __shared__ float sm[8];
    const int lane = threadIdx.x & 31, wave = threadIdx.x >> 5;
    if (lane == 0) sm[wave] = s;
    __syncthreads();
    if (threadIdx.x == 0) {
        float t = 0.0f;
        #pragma unroll
        for (int w = 0; w < 8; ++w) t += sm[w];
        *S = t;
    }
}

// ---------------------------------------------------------------------------
// embed_normalized = avg_new / ((cs_new + eps) * S / (S + C*eps))
// ---------------------------------------------------------------------------
__global__ void k_norm(const float* __restrict__ avg_new, const float* __restrict__ cs_new,
                       const float* __restrict__ S, float* __restrict__ normed) {
    const size_t i = (size_t)blockIdx.x * 256 + threadIdx.x;
    const int c = (int)(i >> 8);   // / DIM
    const float s = *S;
    const float denom = (cs_new[c] + EPSC) * (s / (s + (float)CODES * EPSC));
    normed[i] = avg_new[i] / denom;
}

// ---------------------------------------------------------------------------
extern "C" void kernel_launch(void* const* d_in, const int* in_sizes, int n_in,
                              void* d_out, int out_size, void* d_ws, size_t ws_size,
                              hipStream_t stream) {
    const float* x     = (const float*)d_in[0];  // [16384,256]
    const float* embed = (const float*)d_in[1];  // [8192,256]
    const float* cs    = (const float*)d_in[2];  // [8192]
    const float* avg   = (const float*)d_in[3];  // [8192,256]

    float* out     = (float*)d_out;
    float* quant   = out;                              // 4,194,304
    float* ind_f   = quant + (size_t)N_ROWS * DIM;     // 16,384
    float* normed  = ind_f + N_ROWS;                   // 2,097,152
    float* cs_new  = normed + (size_t)CODES * DIM;     // 8,192
    float* avg_new = cs_new + CODES;                   // 2,097,152

    float* ws_enorm = (float*)d_ws;                                    // 8192 f
    int*   ws_idx   = (int*)((char*)d_ws + (size_t)CODES * 4);         // 16384 i
    float* ws_S     = (float*)((char*)d_ws + (size_t)(CODES + N_ROWS) * 4);

    k_enorm  <<<CODES * 32 / 256, 256, 0, stream>>>(embed, ws_enorm);
    k_init   <<<(CODES * DIM) / 256, 256, 0, stream>>>(cs, avg, cs_new, avg_new);
    k_argmax <<<N_ROWS / MT, 256, 0, stream>>>(x, embed, ws_enorm, ws_idx, ind_f);
    k_scatter<<<N_ROWS, DIM, 0, stream>>>(x, embed, ws_idx, quant, cs_new, avg_new);
    k_sum    <<<1, 256, 0, stream>>>(cs_new, ws_S);
    k_norm   <<<(CODES * DIM) / 256, 256, 0, stream>>>(avg_new, cs_new, ws_S, normed);
}